// GGAHR2HK_13477607375161
// MI455X (gfx1250) — compile-verified
//
#include <hip/hip_runtime.h>
#include <hip/hip_bf16.h>

// ---------------------------------------------------------------------------
// GGAHR2HK pipeline for MI455X (gfx1250, wave32, WMMA).
//  Stage 1: phase table  exp(-2*pi*i * k.S)  -> ws
//  Stage 2: per-edge/per-atom GEMMs with V_WMMA_F32_16X16X4_F32.
//           All LDS operands are zero-padded + spin-deinterleaved at staging
//           time so every WMMA fragment load is an unconditional contiguous
//           pair (ds_load_b64) -- no divergent guards, EXEC stays all-ones.
//  Stage 3: deterministic tile assembly (no atomics): each workgroup owns one
//           (k, a, b) 32x32 (or 26x32) output tile, scans the LDS-staged edge
//           list, and folds the Hermitian  block + block^H  in directly.
// ---------------------------------------------------------------------------

typedef float v2f __attribute__((ext_vector_type(2)));
typedef float v8f __attribute__((ext_vector_type(8)));

#define NORB   13
#define NORB2  26
#define NAUX   32
#define NFEAT  107
#define N_AT   64
#define N_E    1024
#define NK     8

// shell-pair layout (i<=j over shells [1,1,3,3,5], offs [0,1,2,5,8])
__constant__ int   c_fst[15] = {0,1,2,5,8,13,14,17,20,25,34,43,58,67,82};
__constant__ int   c_ist[15] = {0,0,0,0,0, 1, 1, 1, 1, 2, 2, 2, 5, 5, 8};
__constant__ int   c_jst[15] = {0,1,2,5,8, 1, 2, 5, 8, 2, 5, 8, 5, 8, 8};
__constant__ int   c_io [15] = {1,1,1,1,1, 1, 1, 1, 1, 3, 3, 3, 3, 3, 5};
__constant__ int   c_jo [15] = {1,1,3,3,5, 1, 3, 3, 5, 3, 3, 5, 3, 5, 5};
__constant__ float c_fac[15] = {0.5f,1,1,1,1, 0.5f,1,1,1, 0.5f,1,1, 0.5f,1, 0.5f};

// ------------------------------ Stage 1 ------------------------------------
__global__ __launch_bounds__(256)
void phase_kernel(const float* __restrict__ kpoints,
                  const float* __restrict__ shifts,
                  float* __restrict__ phase)   // [NK*N_E][2] (re, im)
{
    int t = blockIdx.x * 256 + threadIdx.x;
    if (t >= NK * N_E) return;
    int k = t >> 10, e = t & (N_E - 1);
    float d = kpoints[k*3+0]*shifts[e*3+0]
            + kpoints[k*3+1]*shifts[e*3+1]
            + kpoints[k*3+2]*shifts[e*3+2];
    float ang = 6.28318530717958647692f * d;   // exp(-2*pi*i*d)
    float sn, cs;
    __sincosf(ang, &sn, &cs);
    phase[2*t]     = cs;
    phase[2*t + 1] = -sn;
}

// ------------------------------ Stage 2 ------------------------------------
// One wave32 per job. Jobs [0, N_E) = edges, [N_E, N_E+N_AT) = onsite atoms.
__global__ __launch_bounds__(32)
void pair_gemm_kernel(const float* __restrict__ hop_feats,
                      const float* __restrict__ ons_feats,
                      const float* __restrict__ R,           // [64][32][26]
                      const int*   __restrict__ edge_index,  // [2][1024]
                      float* __restrict__ hop_tkR,  // [1024][26][32]
                      float* __restrict__ hop_aux,  // [1024][32][32]
                      float* __restrict__ ons_tkR,  // [64][26][32]
                      float* __restrict__ ons_aux)  // [64][32][32]
{
    __shared__ float hs[16 * 16];   // H13 zero-padded to 16x16
    __shared__ float ra[2*32*16];   // A-side R1 spin-split: [s][m][k], k padded
    __shared__ float rb[2*32*16];   // B-side R2 spin-split: [s][q][k], k padded
    __shared__ float cbt[2*32*16];  // C_s transposed: [s][n][m]

    const int lane = threadIdx.x;
    const int job  = blockIdx.x;

    const float *feats, *Rrow, *Rcol;
    float *out_tkR, *out_aux;
    if (job < N_E) {
        int e  = job;
        int ei = edge_index[e];
        int ej = edge_index[N_E + e];
        feats   = hop_feats + (size_t)e * NFEAT;
        Rrow    = R + (size_t)ei * (NAUX * NORB2);
        Rcol    = R + (size_t)ej * (NAUX * NORB2);
        out_tkR = hop_tkR + (size_t)e * (NORB2 * NAUX);
        out_aux = hop_aux + (size_t)e * (NAUX * NAUX);
    } else {
        int n   = job - N_E;
        feats   = ons_feats + (size_t)n * NFEAT;
        Rrow    = R + (size_t)n * (NAUX * NORB2);
        Rcol    = Rrow;
        out_tkR = ons_tkR + (size_t)n * (NORB2 * NAUX);
        out_aux = ons_aux + (size_t)n * (NAUX * NAUX);
    }

    // zero-fill padded operands (k rows 13..15 must be exact zeros)
    for (int t = lane; t < 2 * 32 * 16; t += 32) { ra[t] = 0.f; rb[t] = 0.f; }

    // scatter features into dense zero-padded H13 (upper-tri shell blocks)
    for (int t = lane; t < 256; t += 32) {
        int r = t >> 4, c = t & 15;
        float v = 0.f;
        if (r < NORB && c < NORB) {
            #pragma unroll
            for (int pp = 0; pp < 15; ++pp) {
                int dr = r - c_ist[pp], dc = c - c_jst[pp];
                if (dr >= 0 && dr < c_io[pp] && dc >= 0 && dc < c_jo[pp])
                    v = c_fac[pp] * feats[c_fst[pp] + dr * c_jo[pp] + dc];
            }
        }
        hs[t] = v;
    }
    __syncthreads();

    // spin-deinterleave R into padded [s][row][k] layout
    for (int t = lane; t < NAUX * NORB2; t += 32) {
        int row = t / NORB2;          // aux index (m or q)
        int col = t - row * NORB2;    // 0..25
        int s   = col & 1;
        int kk  = col >> 1;           // 0..12
        int dst = s * (32 * 16) + row * 16 + kk;
        ra[dst] = Rrow[t];
        rb[dst] = Rcol[t];
    }
    __syncthreads();

    const int half = lane >> 4;      // WMMA lane-half (K pair select)
    const int l16  = lane & 15;      // M (A) / N (B,C,D) within tile

    // GEMM1: C_s[16x32] = H13[16x16] @ B_s[16x32], B_s[k][q] = R2[q][2k+s]
    #pragma unroll
    for (int s = 0; s < 2; ++s) {
        #pragma unroll
        for (int tn = 0; tn < 2; ++tn) {
            v8f acc = {0.f,0.f,0.f,0.f,0.f,0.f,0.f,0.f};
            const int q = tn * 16 + l16;
            #pragma unroll
            for (int k0 = 0; k0 < 16; k0 += 4) {
                int ka = k0 + 2 * half;
                v2f a, b;
                a.x = hs[l16 * 16 + ka];
                a.y = hs[l16 * 16 + ka + 1];
                b.x = rb[s * 512 + q * 16 + ka];
                b.y = rb[s * 512 + q * 16 + ka + 1];
                acc = __builtin_amdgcn_wmma_f32_16x16x4_f32(
                          false, a, false, b, (short)0, acc, false, false);
            }
            #pragma unroll
            for (int v = 0; v < 8; ++v)          // store transposed: [n][m]
                cbt[s * 512 + q * 16 + (v + 8 * half)] = acc[v];
        }
    }
    __syncthreads();

    // emit tkR26 with spin interleave: tkR[2i+s][q] = C_s[i][q] = cbt[s][q][i]
    for (int t = lane; t < NORB2 * NAUX; t += 32) {
        int row = t >> 5, q = t & 31;
        out_tkR[t] = cbt[(row & 1) * 512 + q * 16 + (row >> 1)];
    }

    // GEMM2: aux[32x32] = sum_s R1_s[32x16] @ C_s[16x32]
    v8f acc2[2][2];
    #pragma unroll
    for (int tm = 0; tm < 2; ++tm)
        #pragma unroll
        for (int tn = 0; tn < 2; ++tn)
            acc2[tm][tn] = (v8f){0.f,0.f,0.f,0.f,0.f,0.f,0.f,0.f};

    #pragma unroll
    for (int s = 0; s < 2; ++s) {
        #pragma unroll
        for (int k0 = 0; k0 < 16; k0 += 4) {
            int ka = k0 + 2 * half;
            #pragma unroll
            for (int tm = 0; tm < 2; ++tm) {
                v2f a;
                const int m = tm * 16 + l16;
                a.x = ra[s * 512 + m * 16 + ka];
                a.y = ra[s * 512 + m * 16 + ka + 1];
                #pragma unroll
                for (int tn = 0; tn < 2; ++tn) {
                    v2f b;                       // B2_s[k][n] = cbt[s][n][k]
                    const int n = tn * 16 + l16;
                    b.x = cbt[s * 512 + n * 16 + ka];
                    b.y = cbt[s * 512 + n * 16 + ka + 1];
                    acc2[tm][tn] = __builtin_amdgcn_wmma_f32_16x16x4_f32(
                        false, a, false, b, (short)0, acc2[tm][tn], false, false);
                }
            }
        }
    }
    #pragma unroll
    for (int tm = 0; tm < 2; ++tm)
        #pragma unroll
        for (int tn = 0; tn < 2; ++tn)
            #pragma unroll
            for (int v = 0; v < 8; ++v)
                out_aux[(tm * 16 + v + 8 * half) * 32 + tn * 16 + l16] =
                    acc2[tm][tn][v];
}

// ------------------------------ Stage 3 ------------------------------------
// block[k, a*32+p, b*32+q] = sum_{e: ei=a,ej=b} ph(k,e)*aux[e][p][q]
//                          + sum_{e: ei=b,ej=a} conj(ph(k,e))*aux[e][q][p]
//                          + (a==b) [on+LAM][p][q] + [on+LAM][q][p]
// (Hermitian symmetrization folded in; full overwrite, no init needed.)
__global__ __launch_bounds__(256)
void block_kernel(const float* __restrict__ hop_aux,
                  const float* __restrict__ ons_aux,
                  const float* __restrict__ LAM,
                  const float* __restrict__ phase,
                  const int*   __restrict__ edge_index,
                  float* __restrict__ out)       // complex64 [8,2048,2048]
{
    __shared__ int   s_ei[N_E], s_ej[N_E];
    __shared__ float s_pr[N_E], s_pi[N_E];

    const int id  = blockIdx.x;
    const int k   = id >> 12;
    const int a   = (id >> 6) & 63;
    const int b   = id & 63;
    const int tid = threadIdx.x;

    for (int t = tid; t < N_E; t += 256) {
        s_ei[t] = edge_index[t];
        s_ej[t] = edge_index[N_E + t];
        s_pr[t] = phase[2 * (k * N_E + t)];
        s_pi[t] = phase[2 * (k * N_E + t) + 1];
    }
    __syncthreads();

    float ar[4] = {0,0,0,0}, ai[4] = {0,0,0,0};
    int p[4], q[4];
    #pragma unroll
    for (int u = 0; u < 4; ++u) {
        int idx = tid + 256 * u;
        p[u] = idx >> 5; q[u] = idx & 31;
    }

    for (int e = 0; e < N_E; ++e) {
        int ei = s_ei[e], ej = s_ej[e];
        bool m1 = (ei == a) & (ej == b);
        bool m2 = (ei == b) & (ej == a);
        if (m1 | m2) {                                   // uniform branch
            float pr = s_pr[e], pi = s_pi[e];
            const float* M = hop_aux + (size_t)e * (NAUX * NAUX);
            if (m1) {
                #pragma unroll
                for (int u = 0; u < 4; ++u) {
                    float v = M[p[u] * 32 + q[u]];
                    ar[u] += pr * v; ai[u] += pi * v;
                }
            }
            if (m2) {                                    // conj(ph) * M^T
                #pragma unroll
                for (int u = 0; u < 4; ++u) {
                    float w = M[q[u] * 32 + p[u]];
                    ar[u] += pr * w; ai[u] -= pi * w;
                }
            }
        }
    }
    if (a == b) {
        const float* on = ons_aux + (size_t)a * (NAUX * NAUX);
        const float* lm = LAM     + (size_t)a * (NAUX * NAUX);
        #pragma unroll
        for (int u = 0; u < 4; ++u)
            ar[u] += on[p[u]*32+q[u]] + lm[p[u]*32+q[u]]
                   + on[q[u]*32+p[u]] + lm[q[u]*32+p[u]];
    }
    #pragma unroll
    for (int u = 0; u < 4; ++u) {
        size_t row  = (size_t)a * 32 + p[u];
        size_t col  = (size_t)b * 32 + q[u];
        size_t base = 2 * (((size_t)k * 2048 + row) * 2048 + col);
        out[base] = ar[u]; out[base + 1] = ai[u];
    }
}

// tkR[k, a*26+i, b*32+q] = sum_{e: ei=a,ej=b} ph(k,e)*tkR26[e][i][q]
//                        + (a==b) onsite_tkR[a][i][q]
__global__ __launch_bounds__(256)
void tkr_kernel(const float* __restrict__ hop_tkR,
                const float* __restrict__ ons_tkR,
                const float* __restrict__ phase,
                const int*   __restrict__ edge_index,
                float* __restrict__ out)         // complex64 [8,1664,2048]
{
    __shared__ int   s_ei[N_E], s_ej[N_E];
    __shared__ float s_pr[N_E], s_pi[N_E];

    const int id  = blockIdx.x;
    const int k   = id >> 12;
    const int a   = (id >> 6) & 63;
    const int b   = id & 63;
    const int tid = threadIdx.x;

    for (int t = tid; t < N_E; t += 256) {
        s_ei[t] = edge_index[t];
        s_ej[t] = edge_index[N_E + t];
        s_pr[t] = phase[2 * (k * N_E + t)];
        s_pi[t] = phase[2 * (k * N_E + t) + 1];
    }
    __syncthreads();

    float ar[4] = {0,0,0,0}, ai[4] = {0,0,0,0};
    for (int e = 0; e < N_E; ++e) {
        if ((s_ei[e] == a) & (s_ej[e] == b)) {           // uniform branch
            float pr = s_pr[e], pi = s_pi[e];
            const float* M = hop_tkR + (size_t)e * (NORB2 * NAUX);
            #pragma unroll
            for (int u = 0; u < 4; ++u) {
                int idx = tid + 256 * u;
                if (idx < NORB2 * NAUX) {
                    float v = M[idx];
                    ar[u] += pr * v; ai[u] += pi * v;
                }
            }
        }
    }
    if (a == b) {
        const float* on = ons_tkR + (size_t)a * (NORB2 * NAUX);
        #pragma unroll
        for (int u = 0; u < 4; ++u) {
            int idx = tid + 256 * u;
            if (idx < NORB2 * NAUX) ar[u] += on[idx];
        }
    }
    #pragma unroll
    for (int u = 0; u < 4; ++u) {
        int idx = tid + 256 * u;
        if (idx < NORB2 * NAUX) {
            size_t row  = (size_t)a * NORB2 + (idx >> 5);
            size_t col  = (size_t)b * NAUX  + (idx & 31);
            size_t base = 2 * (((size_t)k * 1664 + row) * 2048 + col);
            out[base] = ar[u]; out[base + 1] = ai[u];
        }
    }
}

// ------------------------------ launch -------------------------------------
extern "C" void kernel_launch(void* const* d_in, const int* in_sizes, int n_in,
                              void* d_out, int out_size, void* d_ws, size_t ws_size,
                              hipStream_t stream)
{
    const float* hop_feats = (const float*)d_in[0];  // [1024,107]
    const float* ons_feats = (const float*)d_in[1];  // [64,107]
    const float* R         = (const float*)d_in[2];  // [64,32,26]
    const float* LAM       = (const float*)d_in[3];  // [64,32,32]
    const float* kpoints   = (const float*)d_in[4];  // [8,3]
    const float* shifts    = (const float*)d_in[5];  // [1024,3]
    const int*   eidx      = (const int*)  d_in[6];  // [2,1024]

    float* ws = (float*)d_ws;
    float* hop_tkR = ws;                                   // 1024*832
    float* hop_aux = hop_tkR + (size_t)N_E * NORB2 * NAUX; // 1024*1024
    float* ons_tkR = hop_aux + (size_t)N_E * NAUX * NAUX;  // 64*832
    float* ons_aux = ons_tkR + (size_t)N_AT * NORB2 * NAUX;// 64*1024
    float* phase   = ons_aux + (size_t)N_AT * NAUX * NAUX; // 8*1024*2

    float* out_block = (float*)d_out;                              // 8*2048*2048 cplx
    float* out_tkR   = out_block + (size_t)NK * 2048 * 2048 * 2;   // 8*1664*2048 cplx

    phase_kernel<<<(NK * N_E + 255) / 256, 256, 0, stream>>>(kpoints, shifts, phase);

    pair_gemm_kernel<<<N_E + N_AT, 32, 0, stream>>>(
        hop_feats, ons_feats, R, eidx, hop_tkR, hop_aux, ons_tkR, ons_aux);

    block_kernel<<<NK * N_AT * N_AT, 256, 0, stream>>>(
        hop_aux, ons_aux, LAM, phase, eidx, out_block);

    tkr_kernel<<<NK * N_AT * N_AT, 256, 0, stream>>>(
        hop_tkR, ons_tkR, phase, eidx, out_tkR);
}